// DynamicsParameter_48077863911582
// MI455X (gfx1250) — compile-verified
//
#include <hip/hip_runtime.h>
#include <hip/hip_bf16.h>

// ---------------------------------------------------------------------------
// Problem: B=128, T=1024, K=16, n=16, m=8, p=32 (fp32 everywhere).
// Memory-bound (~340MB traffic ~15us @ 23.3TB/s). Scan is sequential in T,
// parallel over B -> one wave32 per batch, shuffle reductions, register
// double-buffering + global_prefetch. Mixing einsums are GEMMs (131072x16 @
// 16x{256,128}) -> V_WMMA_F32_16X16X4_F32, 4 chained K=4 WMMAs per tile.
// ---------------------------------------------------------------------------

typedef float v2f __attribute__((ext_vector_type(2)));
typedef float v8f __attribute__((ext_vector_type(8)));

#define TAU_INV 2.0f   // 1/0.5

static constexpr int Bn = 128;
static constexpr int Tn = 1024;
static constexpr int Kn = 16;

// float offsets into d_out (return order: A_seq, B_seq, C_seq, log_q, log_p)
static constexpr size_t OFF_A  = 0;                       // 128*1024*256
static constexpr size_t OFF_B  = 33554432;                // + 128*1024*128
static constexpr size_t OFF_C  = 50331648;                // + 128*1*32*16
static constexpr size_t OFF_LQ = 50397184;                // + 128*1024
static constexpr size_t OFF_LP = 50528256;                // + 128*1024

__device__ __forceinline__ float bsum16(float v) {
    v += __shfl_xor(v, 8, 32);
    v += __shfl_xor(v, 4, 32);
    v += __shfl_xor(v, 2, 32);
    v += __shfl_xor(v, 1, 32);
    return v;
}
__device__ __forceinline__ float bmax16(float v) {
    v = fmaxf(v, __shfl_xor(v, 8, 32));
    v = fmaxf(v, __shfl_xor(v, 4, 32));
    v = fmaxf(v, __shfl_xor(v, 2, 32));
    v = fmaxf(v, __shfl_xor(v, 1, 32));
    return v;
}

// -------------------- Phase 1: sequential regime scan ----------------------
// One wave32 per batch. Lane j (j = lane&15) owns regime j; lanes 16..31
// mirror lanes 0..15 so EXEC stays full for shuffles. Per step: 16 coalesced
// 64B column loads (next-step registers prefilled), softmax/log-softmax via
// butterfly reductions, 16 shuffles to rebuild the dense y vector.
__global__ __launch_bounds__(32) void regime_scan_kernel(
    const float* __restrict__ logits,   // (B,T,16,16)
    const float* __restrict__ gumbel,   // (B,T,16)
    const float* __restrict__ transP,   // (16,16)
    float* __restrict__ y_seq,          // ws: (B,T,16)
    float* __restrict__ log_q,          // (B,T)
    float* __restrict__ log_p)          // (B,T)
{
    const int b    = blockIdx.x;
    const int lane = threadIdx.x;
    const int j    = lane & 15;

    // log P row j (for lp bilinear form), loaded once
    float logPr[16];
#pragma unroll
    for (int k = 0; k < 16; ++k) logPr[k] = __logf(transP[j * 16 + k]);

    float yfull[16];
#pragma unroll
    for (int k = 0; k < 16; ++k) yfull[k] = 0.0625f;
    float yprev = 0.0625f;

    const float* Lb = logits + (size_t)b * Tn * 256;
    const float* Gb = gumbel + (size_t)b * Tn * 16;
    float* Yb = y_seq + (size_t)b * Tn * 16;

    // preload t=0 column j of the 16x16 logit tile
    float col[16];
#pragma unroll
    for (int k = 0; k < 16; ++k) col[k] = Lb[k * 16 + j];
    float g = Gb[j];

    for (int t = 0; t < Tn; ++t) {
        // L2 prefetch a few tiles ahead (1KB tile, 32B per lane)
        if (t + 4 < Tn)
            __builtin_prefetch(Lb + (size_t)(t + 4) * 256 + lane * 8, 0, 0);

        // register double-buffer: issue next step's loads now
        float coln[16] = {};
        float gn = 0.0f;
        if (t + 1 < Tn) {
            const float* Ln = Lb + (size_t)(t + 1) * 256;
#pragma unroll
            for (int k = 0; k < 16; ++k) coln[k] = Ln[k * 16 + j];
            gn = Gb[(t + 1) * 16 + j];
        }

        // l_j = sum_k y_prev[k] * L[k][j]
        float l = 0.0f;
#pragma unroll
        for (int k = 0; k < 16; ++k) l = __fmaf_rn(yfull[k], col[k], l);

        // gumbel-softmax y = softmax((l+g)/tau)
        float z  = (l + g) * TAU_INV;
        float zm = bmax16(z);
        float e  = __expf(z - zm);
        float es = bsum16(e);
        float y  = e / es;

        // lq = sum_j y_j * log_softmax(l)_j
        float lm  = bmax16(l);
        float le  = __expf(l - lm);
        float les = bsum16(le);
        float lsm = l - lm - __logf(les);
        float lq  = bsum16(y * lsm);

        // dense new y via shuffles
        float ynew[16];
#pragma unroll
        for (int k = 0; k < 16; ++k) ynew[k] = __shfl(y, k, 32);

        // lp = y_prev^T logP y  (t=0 uses -log K)
        float s = 0.0f;
#pragma unroll
        for (int k = 0; k < 16; ++k) s = __fmaf_rn(logPr[k], ynew[k], s);
        float lp = bsum16(yprev * s);
        if (t == 0) lp = -__logf(16.0f);

        if (lane < 16) Yb[t * 16 + j] = y;
        if (lane == 0) {
            log_q[(size_t)b * Tn + t] = lq;
            log_p[(size_t)b * Tn + t] = lp;
        }

        yprev = y;
#pragma unroll
        for (int k = 0; k < 16; ++k) { yfull[k] = ynew[k]; col[k] = coln[k]; }
        g = gn;
    }
}

// -------------------- Phase 2: mixing GEMMs via f32 WMMA -------------------
// A_seq = Y(131072x16) @ A(16x256); B_seq = Y @ Bm(16x128).
// One wave per 16-row M-tile; each 16x16 output tile = 4 chained
// V_WMMA_F32_16X16X4_F32 (K = 4q..4q+3). Fragment layouts per ISA 7.12.2:
//   A 16x4 f32 : lane m (m=lane&15) holds {K=4q+2h, 4q+2h+1}, h=lane>>4
//   B 4x16 f32 : lane n holds column n, same K split
//   C/D 16x16  : VGPR v -> M = v + 8h, N = lane&15
__global__ __launch_bounds__(256) void mix_wmma_kernel(
    const float* __restrict__ y_seq,   // (BT,16)
    const float* __restrict__ Ap,      // (16,256)
    const float* __restrict__ Bp,      // (16,128)
    float* __restrict__ Aout,          // (BT,256)
    float* __restrict__ Bout)          // (BT,128)
{
    const int lane = threadIdx.x & 31;
    const int wave = threadIdx.x >> 5;
    const int tile = blockIdx.x * 8 + wave;     // 8192 M-tiles total
    const int n    = lane & 15;
    const int h    = lane >> 4;
    const int rowbase = tile * 16;

    // A-operand fragments (the y rows), reused for all 24 N-tiles
    v2f afrag[4];
    const float* yrow = y_seq + (size_t)(rowbase + n) * 16;
#pragma unroll
    for (int q = 0; q < 4; ++q) {
        const float2 v = *(const float2*)(yrow + 4 * q + 2 * h);
        afrag[q].x = v.x;
        afrag[q].y = v.y;
    }

    // ---- A_seq: 16 N-tiles of width 16 over N=256 ----
#pragma unroll
    for (int nt = 0; nt < 16; ++nt) {
        v8f c = {};
#pragma unroll
        for (int q = 0; q < 4; ++q) {
            v2f bfrag;
            bfrag.x = Ap[(4 * q + 2 * h) * 256 + nt * 16 + n];
            bfrag.y = Ap[(4 * q + 2 * h + 1) * 256 + nt * 16 + n];
            c = __builtin_amdgcn_wmma_f32_16x16x4_f32(
                    false, afrag[q], false, bfrag, (short)0, c, false, false);
        }
#pragma unroll
        for (int v = 0; v < 8; ++v)
            Aout[(size_t)(rowbase + v + 8 * h) * 256 + nt * 16 + n] = c[v];
    }

    // ---- B_seq: 8 N-tiles of width 16 over N=128 ----
#pragma unroll
    for (int nt = 0; nt < 8; ++nt) {
        v8f c = {};
#pragma unroll
        for (int q = 0; q < 4; ++q) {
            v2f bfrag;
            bfrag.x = Bp[(4 * q + 2 * h) * 128 + nt * 16 + n];
            bfrag.y = Bp[(4 * q + 2 * h + 1) * 128 + nt * 16 + n];
            c = __builtin_amdgcn_wmma_f32_16x16x4_f32(
                    false, afrag[q], false, bfrag, (short)0, c, false, false);
        }
#pragma unroll
        for (int v = 0; v < 8; ++v)
            Bout[(size_t)(rowbase + v + 8 * h) * 128 + nt * 16 + n] = c[v];
    }
}

// -------------------- Phase 3: C_seq broadcast -----------------------------
__global__ __launch_bounds__(256) void cseq_kernel(const float* __restrict__ Cm,
                                                   float* __restrict__ Cout) {
    const int i = blockIdx.x * 256 + threadIdx.x;   // 65536 = 128 * 512
    Cout[i] = Cm[i & 511];
}

extern "C" void kernel_launch(void* const* d_in, const int* in_sizes, int n_in,
                              void* d_out, int out_size, void* d_ws, size_t ws_size,
                              hipStream_t stream) {
    const float* logits = (const float*)d_in[0];
    const float* gumbel = (const float*)d_in[1];
    const float* Ap     = (const float*)d_in[2];
    const float* Bp     = (const float*)d_in[3];
    const float* Cm     = (const float*)d_in[4];
    const float* transP = (const float*)d_in[5];

    float* out  = (float*)d_out;
    float* yseq = (float*)d_ws;   // 8 MB scratch for y_seq

    regime_scan_kernel<<<Bn, 32, 0, stream>>>(logits, gumbel, transP,
                                              yseq, out + OFF_LQ, out + OFF_LP);
    mix_wmma_kernel<<<(Bn * Tn / 16) / 8, 256, 0, stream>>>(
        yseq, Ap, Bp, out + OFF_A, out + OFF_B);
    cseq_kernel<<<256, 256, 0, stream>>>(Cm, out + OFF_C);
}